// GumbelVectorQuantizer_42838003810498
// MI455X (gfx1250) — compile-verified
//
#include <hip/hip_runtime.h>
#include <math.h>

typedef __attribute__((ext_vector_type(2))) float v2f;
typedef __attribute__((ext_vector_type(4))) float v4f;
typedef __attribute__((ext_vector_type(8))) float v8f;

#define BM 256
#define BN 64
#define BK 32
#define KP   (BK / 2)  // 16 paired-k rows
#define LDAP 257       // Ap stride in float2 units (padded)
#define LDBP 68        // Bp stride in float2 units (padded)

// logits[M,N] = hidden[M,K] @ W[K,N] + b  via V_WMMA_F32_16X16X4_F32.
// M=32768, K=512, N=640. Block tile 256x64, K-chunk 32, 256 threads = 8 waves;
// each wave owns 32 rows x 64 cols (8 f32 accumulator fragments).
// LDS holds A/B pre-paired as float2 over (k, k+1) so every WMMA operand is a
// single aligned ds_load_b64 straight into the register pair WMMA consumes.
__global__ __launch_bounds__(256) void gemm_logits_kernel(
    const float* __restrict__ A, const float* __restrict__ W,
    const float* __restrict__ bias, float* __restrict__ out,
    int M, int N, int K)
{
  __shared__ v2f Ap[KP * LDAP];   // Ap[k2][m] = {A[2k2][m], A[2k2+1][m]}
  __shared__ v2f Bp[KP * LDBP];   // Bp[k2][n] = {B[2k2][n], B[2k2+1][n]}

  const int t    = threadIdx.x;
  const int lane = t & 31;
  const int wave = t >> 5;
  const int gm0  = blockIdx.x * BM;
  const int gn0  = blockIdx.y * BN;
  const int h    = lane >> 4;   // half-wave group
  const int lm   = lane & 15;
  const int wm   = wave * 32;

  v8f acc[2][4];
  v8f zero = {0.f,0.f,0.f,0.f,0.f,0.f,0.f,0.f};
  for (int mt = 0; mt < 2; ++mt)
    for (int nt = 0; nt < 4; ++nt)
      acc[mt][nt] = zero;

  for (int kc = 0; kc < K; kc += BK) {
    // ---- stage A chunk (256 rows x 32 k), coalesced 128B per row octet ----
    for (int i = 0; i < 8; ++i) {
      int idx = i * 256 + t;
      int m   = idx >> 3;
      int k4  = (idx & 7) << 2;           // first of 4 consecutive k
      v4f a4 = *(const v4f*)(A + (size_t)(gm0 + m) * K + kc + k4);
      v2f p0 = {a4.x, a4.y};
      v2f p1 = {a4.z, a4.w};
      Ap[((k4 >> 1) + 0) * LDAP + m] = p0;
      Ap[((k4 >> 1) + 1) * LDAP + m] = p1;
    }
    // ---- stage B chunk (32 k x 64 n): read rows k,k+1 and interleave ----
    {
      int k2 = t >> 4;                    // 0..15
      int n4 = (t & 15) << 2;             // 0..60
      const float* wr = W + (size_t)(kc + 2 * k2) * N + gn0 + n4;
      v4f r0 = *(const v4f*)(wr);
      v4f r1 = *(const v4f*)(wr + N);
      v4f q0 = {r0.x, r1.x, r0.y, r1.y};
      v4f q1 = {r0.z, r1.z, r0.w, r1.w};
      *(v4f*)(&Bp[k2 * LDBP + n4 + 0]) = q0;
      *(v4f*)(&Bp[k2 * LDBP + n4 + 2]) = q1;
    }
    // prefetch next A chunk (one 128B cacheline per thread = full next tile)
    if (kc + BK < K)
      __builtin_prefetch(A + (size_t)(gm0 + t) * K + kc + BK, 0, 1);
    __syncthreads();

    // ---- compute: 8 k-steps of 4, 8 WMMAs per step ----
    for (int ks2 = 0; ks2 < KP; ks2 += 2) {
      const int kk = ks2 + h;             // paired-k row for this half-wave
      v2f a0 = Ap[kk * LDAP + wm + lm];
      v2f a1 = Ap[kk * LDAP + wm + 16 + lm];
      v2f b0 = Bp[kk * LDBP +  0 + lm];
      v2f b1 = Bp[kk * LDBP + 16 + lm];
      v2f b2 = Bp[kk * LDBP + 32 + lm];
      v2f b3 = Bp[kk * LDBP + 48 + lm];
      acc[0][0] = __builtin_amdgcn_wmma_f32_16x16x4_f32(false, a0, false, b0, (short)0, acc[0][0], false, false);
      acc[1][0] = __builtin_amdgcn_wmma_f32_16x16x4_f32(false, a1, false, b0, (short)0, acc[1][0], false, false);
      acc[0][1] = __builtin_amdgcn_wmma_f32_16x16x4_f32(false, a0, false, b1, (short)0, acc[0][1], false, false);
      acc[1][1] = __builtin_amdgcn_wmma_f32_16x16x4_f32(false, a1, false, b1, (short)0, acc[1][1], false, false);
      acc[0][2] = __builtin_amdgcn_wmma_f32_16x16x4_f32(false, a0, false, b2, (short)0, acc[0][2], false, false);
      acc[1][2] = __builtin_amdgcn_wmma_f32_16x16x4_f32(false, a1, false, b2, (short)0, acc[1][2], false, false);
      acc[0][3] = __builtin_amdgcn_wmma_f32_16x16x4_f32(false, a0, false, b3, (short)0, acc[0][3], false, false);
      acc[1][3] = __builtin_amdgcn_wmma_f32_16x16x4_f32(false, a1, false, b3, (short)0, acc[1][3], false, false);
    }
    __syncthreads();
  }

  // ---- epilogue: bias + store (coalesced b32 across lanes) ----
  for (int nt = 0; nt < 4; ++nt) {
    int col  = gn0 + nt * 16 + lm;
    float bv = bias[col];
    for (int mt = 0; mt < 2; ++mt) {
      for (int r = 0; r < 8; ++r) {
        int row = gm0 + wm + mt * 16 + r + 8 * h;
        out[(size_t)row * N + col] = acc[mt][nt][r] + bv;
      }
    }
  }
}

// One wave per 16 rows of one codebook group; V=320 -> 10 elems/lane (wave32).
__global__ __launch_bounds__(256) void gumbel_kernel(
    const float* __restrict__ logits,    // [32768, 640]
    const int*   __restrict__ lengths,   // [16]
    const float* __restrict__ code_book, // [2,320,384]
    const float* __restrict__ gumbel_u,  // [65536, 320]
    float* __restrict__ code_vectors,    // [32768, 768]
    float* __restrict__ perp_acc)        // [640]
{
  const int lane = threadIdx.x & 31;
  const int wgl  = blockIdx.x * 8 + (threadIdx.x >> 5);  // 0..4095
  const int g    = wgl >> 11;                            // 0..1
  const int n0   = (wgl & 2047) * 16;

  float perp[10];
  for (int j = 0; j < 10; ++j) perp[j] = 0.f;

  for (int r = 0; r < 16; ++r) {
    const int n     = n0 + r;
    const int b_idx = n >> 11;    // / 2048
    const int l_idx = n & 2047;
    const bool masked = l_idx < lengths[b_idx];

    const float* lrow = logits   + (size_t)n * 640 + g * 320;
    const float* urow = gumbel_u + ((size_t)n * 2 + g) * 320;

    float l[10], z[10];
    float m1 = -INFINITY, m2 = -INFINITY;
    int   bi = 0;
    for (int j = 0; j < 10; ++j) {
      const int v  = lane + 32 * j;
      const float lv = lrow[v];
      const float u  = urow[v];
      const float gn = -__logf(-__logf(u));
      const float zv = (lv + gn) * 0.5f;   // /TAU, TAU=2
      l[j] = lv; z[j] = zv;
      m1 = fmaxf(m1, lv);
      if (zv > m2) { m2 = zv; bi = v; }    // first max kept (v increasing)
    }
    for (int off = 16; off; off >>= 1)
      m1 = fmaxf(m1, __shfl_xor(m1, off, 32));
    for (int off = 16; off; off >>= 1) {
      float ov = __shfl_xor(m2, off, 32);
      int   oi = __shfl_xor(bi, off, 32);
      if (ov > m2 || (ov == m2 && oi < bi)) { m2 = ov; bi = oi; }
    }

    float s1 = 0.f, s2 = 0.f;
    float e1[10];
    for (int j = 0; j < 10; ++j) {
      e1[j] = __expf(l[j] - m1);
      s1 += e1[j];
      s2 += __expf(z[j] - m2);
    }
    for (int off = 16; off; off >>= 1) s1 += __shfl_xor(s1, off, 32);
    for (int off = 16; off; off >>= 1) s2 += __shfl_xor(s2, off, 32);

    if (masked) {
      const float inv = 1.0f / s1;
      for (int j = 0; j < 10; ++j) perp[j] += e1[j] * inv;
    }

    // straight-through: probs == one_hot except fl((1+s)-s) at argmax
    const float s     = 1.0f / s2;          // y_soft at argmax (exp(0)/sum)
    const float scale = (1.0f + s) - s;
    const float* cb = code_book + ((size_t)g * 320 + bi) * 384;
    float* orow = code_vectors + (size_t)n * 768 + g * 384;
    for (int e = lane; e < 384; e += 32) orow[e] = scale * cb[e];
  }

  for (int j = 0; j < 10; ++j)
    atomicAdd(&perp_acc[g * 320 + lane + 32 * j], perp[j]);
}

__global__ void init_kernel(const int* __restrict__ lengths,
                            float* __restrict__ perp_acc,
                            float* __restrict__ nv)
{
  const int t = threadIdx.x;
  for (int i = t; i < 640; i += blockDim.x) perp_acc[i] = 0.f;
  if (t == 0) {
    int s = 0;
    for (int i = 0; i < 16; ++i) s += lengths[i];
    *nv = (float)s;
  }
}

__global__ void finalize_kernel(const float* __restrict__ perp_acc,
                                const float* __restrict__ nv,
                                float* __restrict__ out)
{
  const int i = blockIdx.x * blockDim.x + threadIdx.x;
  if (i < 640) out[i] = perp_acc[i] / (*nv);
}

extern "C" void kernel_launch(void* const* d_in, const int* in_sizes, int n_in,
                              void* d_out, int out_size, void* d_ws, size_t ws_size,
                              hipStream_t stream)
{
  const float* hidden   = (const float*)d_in[0];
  const int*   lengths  = (const int*)  d_in[1];
  const float* W        = (const float*)d_in[2];
  const float* b        = (const float*)d_in[3];
  const float* codebook = (const float*)d_in[4];
  const float* gumbel_u = (const float*)d_in[5];

  float* out      = (float*)d_out;
  float* logits   = (float*)d_ws;                  // 32768*640 floats (84 MB)
  float* perp_acc = logits + (size_t)32768 * 640;  // 640 floats
  float* nv       = perp_acc + 640;                // 1 float

  float* code_vectors = out;                       // 32768*768 floats
  float* perp_out     = out + (size_t)32768 * 768; // 640 floats

  init_kernel<<<1, 128, 0, stream>>>(lengths, perp_acc, nv);

  dim3 ggrid(32768 / BM, 640 / BN);                // 128 x 10
  gemm_logits_kernel<<<ggrid, 256, 0, stream>>>(hidden, W, b, logits,
                                                32768, 640, 512);

  gumbel_kernel<<<512, 256, 0, stream>>>(logits, lengths, codebook, gumbel_u,
                                         code_vectors, perp_acc);

  finalize_kernel<<<2, 320, 0, stream>>>(perp_acc, nv, perp_out);
}